// GMVAE_90400471646808
// MI455X (gfx1250) — compile-verified
//
#include <hip/hip_runtime.h>
#include <cmath>

typedef __attribute__((ext_vector_type(2))) float v2f;
typedef __attribute__((ext_vector_type(8))) float v8f;

#define B_    16
#define S_    256
#define D_    512
#define NH_   4
#define DH_   128
#define L_    64
#define K_    20
#define MC_   10
#define BS_   (B_*S_)      // 4096
#define NALL_ (S_*MC_)     // 2560
#define ROWS_ (BS_*MC_)    // 40960
#define LOG_NORM_ (-0.91893853320467274178f)
#define C0_ (64.0f*LOG_NORM_)

// ---------------- WMMA helper: D = A(16x4,f32) * B(4x16,f32) + C ----------------
__device__ inline v8f wmma4(v2f a, v2f b, v8f c) {
  return __builtin_amdgcn_wmma_f32_16x16x4_f32(false, a, false, b, (short)0, c, false, false);
}
__device__ inline float wsum(float v){ for(int o=16;o;o>>=1) v += __shfl_xor(v,o,32); return v; }
__device__ inline float wmax(float v){ for(int o=16;o;o>>=1) v = fmaxf(v,__shfl_xor(v,o,32)); return v; }

// ---------------- generic GEMM: C = op(A@W + bias), one wave per 16x16 tile ------
// mode 0: C = acc+bias ; mode 2: C = Res + relu(acc+bias)
__global__ void gemm_kernel(const float* __restrict__ A, const float* __restrict__ Bm,
                            const float* __restrict__ bias, const float* __restrict__ Res,
                            float* __restrict__ C, int N, int Kd, int mode) {
  const int lane = threadIdx.x;
  const int n0 = blockIdx.x*16, m0 = blockIdx.y*16;
  const int mrow = lane&15, kh = (lane>>4)*2, ncol = lane&15, hl = lane>>4;
  const float* ar = A + (size_t)(m0+mrow)*Kd;
  v8f acc = {};
  for (int k0=0;k0<Kd;k0+=4){
    v2f a,bb;
    a.x  = ar[k0+kh];                      a.y  = ar[k0+kh+1];
    bb.x = Bm[(size_t)(k0+kh)*N + n0+ncol]; bb.y = Bm[(size_t)(k0+kh+1)*N + n0+ncol];
    acc = wmma4(a,bb,acc);
  }
  float bv = bias ? bias[n0+ncol] : 0.f;
  for (int r=0;r<8;++r){
    int row = m0 + r + 8*hl;
    float v = acc[r] + bv;
    if (mode==2) v = Res[(size_t)row*N + n0+ncol] + fmaxf(v,0.f);
    C[(size_t)row*N + n0+ncol] = v;
  }
}

// ---------------- attention scores: Att = scale * q @ k^T per (b,h) --------------
__global__ void attn_scores_kernel(const float* __restrict__ Q, const float* __restrict__ Km,
                                   float* __restrict__ Att, float scale) {
  const int lane = threadIdx.x;
  const int bh = blockIdx.z, b = bh>>2, h = bh&3;
  const int n0 = blockIdx.x*16, m0 = blockIdx.y*16;
  const int mrow = lane&15, kh = (lane>>4)*2, ncol = lane&15, hl = lane>>4;
  const float* qr = Q  + (size_t)(b*S_+m0+mrow)*D_ + h*DH_;
  const float* kr = Km + (size_t)(b*S_+n0+ncol)*D_ + h*DH_;
  v8f acc = {};
  for (int k0=0;k0<DH_;k0+=4){
    v2f a,bb;
    a.x = qr[k0+kh];  a.y = qr[k0+kh+1];
    bb.x = kr[k0+kh]; bb.y = kr[k0+kh+1];
    acc = wmma4(a,bb,acc);
  }
  float* o = Att + (size_t)bh*S_*S_;
  for (int r=0;r<8;++r)
    o[(size_t)(m0+r+8*hl)*S_ + n0+ncol] = acc[r]*scale;
}

// ---------------- row softmax over 256 cols, one wave per row --------------------
__global__ __launch_bounds__(256) void softmax_kernel(float* __restrict__ Att) {
  const int lane = threadIdx.x&31, wave = threadIdx.x>>5;
  float* p = Att + ((size_t)blockIdx.x*8 + wave)*S_;
  float v[8]; float mx = -3.0e38f;
  for (int j=0;j<8;++j){ v[j] = p[lane+32*j]; mx = fmaxf(mx, v[j]); }
  mx = wmax(mx);
  float s = 0.f;
  for (int j=0;j<8;++j){ v[j] = expf(v[j]-mx); s += v[j]; }
  s = wsum(s);
  float inv = 1.f/s;
  for (int j=0;j<8;++j) p[lane+32*j] = v[j]*inv;
}

// ---------------- O = q + Att @ v per (b,h): C preloaded with q ------------------
__global__ void attn_v_kernel(const float* __restrict__ Att, const float* __restrict__ Q,
                              const float* __restrict__ V, float* __restrict__ O) {
  const int lane = threadIdx.x;
  const int bh = blockIdx.z, b = bh>>2, h = bh&3;
  const int n0 = blockIdx.x*16, m0 = blockIdx.y*16;
  const int mrow = lane&15, kh = (lane>>4)*2, ncol = lane&15, hl = lane>>4;
  const float* ar = Att + (size_t)bh*S_*S_ + (size_t)(m0+mrow)*S_;
  v8f acc;
  for (int r=0;r<8;++r)
    acc[r] = Q[(size_t)(b*S_+m0+r+8*hl)*D_ + h*DH_ + n0+ncol];
  for (int k0=0;k0<S_;k0+=4){
    v2f a,bb;
    a.x = ar[k0+kh]; a.y = ar[k0+kh+1];
    bb.x = V[(size_t)(b*S_+k0+kh)*D_   + h*DH_ + n0+ncol];
    bb.y = V[(size_t)(b*S_+k0+kh+1)*D_ + h*DH_ + n0+ncol];
    acc = wmma4(a,bb,acc);
  }
  for (int r=0;r<8;++r)
    O[(size_t)(b*S_+m0+r+8*hl)*D_ + h*DH_ + n0+ncol] = acc[r];
}

// ---------------- z = mean + eps*std ; x2 = sum z^2 (wave per (bs,mc)) ----------
__global__ __launch_bounds__(256) void z_kernel(const float* __restrict__ stats,
                                                const float* __restrict__ eps,
                                                float* __restrict__ Z, float* __restrict__ x2) {
  const int lane = threadIdx.x&31, wave = threadIdx.x>>5;
  const int gw = blockIdx.x*8 + wave;
  const int bs = gw / MC_;
  float m0  = stats[(size_t)bs*128 + lane];
  float m1  = stats[(size_t)bs*128 + lane+32];
  float lv0 = stats[(size_t)bs*128 + 64 + lane];
  float lv1 = stats[(size_t)bs*128 + 96 + lane];
  float e0 = eps[(size_t)gw*64 + lane], e1 = eps[(size_t)gw*64 + lane+32];
  float z0 = m0 + e0*expf(0.5f*lv0);
  float z1 = m1 + e1*expf(0.5f*lv1);
  Z[(size_t)gw*64 + lane]    = z0;
  Z[(size_t)gw*64 + lane+32] = z1;
  float p = wsum(z0*z0 + z1*z1);
  if (lane==0) x2[gw] = p;
}

__global__ void mu_init_kernel(const float* __restrict__ Z, const int* __restrict__ idx,
                               float* __restrict__ mu, float* __restrict__ pi) {
  const int b = blockIdx.x, k = blockIdx.y, t = threadIdx.x;
  const int n = idx[b*K_+k];
  mu[(size_t)(b*K_+k)*64 + t] = Z[(size_t)(b*NALL_ + n)*64 + t];
  if (t==0) pi[b*K_+k] = 1.0f/(float)K_;
}

__global__ void em_prep_kernel(const float* __restrict__ mu, const float* __restrict__ pi,
                               float* __restrict__ m2, float* __restrict__ lp) {
  const int b = blockIdx.x, k = threadIdx.x;
  if (k < K_) {
    const float* mr = mu + (size_t)(b*K_+k)*64;
    float s = 0.f;
    for (int l=0;l<64;++l) s += mr[l]*mr[l];
    m2[b*K_+k] = s;
    lp[b*K_+k] = logf(pi[b*K_+k]);
  }
}

// ---------------- EM E-step + accumulation (wave per 32 rows) --------------------
__global__ __launch_bounds__(256) void em_accum_kernel(
    const float* __restrict__ Z, const float* __restrict__ x2,
    const float* __restrict__ mu, const float* __restrict__ m2, const float* __restrict__ lp,
    float* __restrict__ muAcc, float* __restrict__ NkAcc) {
  __shared__ float sMu[K_*64];
  __shared__ float sM2[K_], sLp[K_];
  const int b = blockIdx.x;
  const int lane = threadIdx.x&31, wave = threadIdx.x>>5;
  for (int i=threadIdx.x; i<K_*64; i+=256) sMu[i] = mu[b*K_*64 + i];
  if (threadIdx.x < K_) { sM2[threadIdx.x] = m2[b*K_+threadIdx.x]; sLp[threadIdx.x] = lp[b*K_+threadIdx.x]; }
  __syncthreads();
  float aN[K_], aM0[K_], aM1[K_];
  #pragma unroll
  for (int k=0;k<K_;++k){ aN[k]=0.f; aM0[k]=0.f; aM1[k]=0.f; }
  const int nbase = (blockIdx.y*8 + wave)*32;
  for (int t=0;t<32;++t) {
    const int n = nbase + t;
    const float* zr = Z + (size_t)(b*NALL_ + n)*64;
    float z0 = zr[lane], z1 = zr[lane+32];
    float ll[K_];
    #pragma unroll
    for (int k=0;k<K_;++k) {
      float p = z0*sMu[k*64+lane] + z1*sMu[k*64+lane+32];
      ll[k] = wsum(p);
    }
    float xv = x2[b*NALL_ + n];
    float mx = -3.0e38f;
    #pragma unroll
    for (int k=0;k<K_;++k) {
      ll[k] = -0.5f*(xv - 2.f*ll[k] + sM2[k]) + C0_ + sLp[k];
      mx = fmaxf(mx, ll[k]);
    }
    float ssum = 0.f;
    #pragma unroll
    for (int k=0;k<K_;++k) { ll[k] = expf(ll[k]-mx); ssum += ll[k]; }
    float inv = 1.f/ssum;
    #pragma unroll
    for (int k=0;k<K_;++k) {
      float p = ll[k]*inv;
      aN[k] += p; aM0[k] += p*z0; aM1[k] += p*z1;
    }
  }
  #pragma unroll
  for (int k=0;k<K_;++k) {
    atomicAdd(&muAcc[(size_t)(b*K_+k)*64 + lane],      aM0[k]);
    atomicAdd(&muAcc[(size_t)(b*K_+k)*64 + lane + 32], aM1[k]);
  }
  if (lane==0) {
    #pragma unroll
    for (int k=0;k<K_;++k) atomicAdd(&NkAcc[b*K_+k], aN[k]);
  }
}

__global__ void em_final_kernel(const float* __restrict__ muAcc, const float* __restrict__ NkAcc,
                                float* __restrict__ mu, float* __restrict__ pi) {
  __shared__ float sNk[K_]; __shared__ float sSum;
  const int b = blockIdx.x, t = threadIdx.x;   // 64 threads
  if (t < K_) sNk[t] = NkAcc[b*K_+t];
  __syncthreads();
  if (t==0){ float s=0.f; for(int k=0;k<K_;++k) s += sNk[k]; sSum = s; }
  __syncthreads();
  if (t < K_) pi[b*K_+t] = sNk[t]/sSum;
  for (int k=0;k<K_;++k)
    mu[(size_t)(b*K_+k)*64 + t] = muAcc[(size_t)(b*K_+k)*64 + t] / sNk[k];
}

// ---------------- fused 3-layer decoder + rec loss, 16 rows per WG ---------------
#define LDSW 516
__global__ __launch_bounds__(256) void decoder_kernel(
    const float* __restrict__ Z, const float* __restrict__ Hin,
    const float* __restrict__ w1, const float* __restrict__ b1,
    const float* __restrict__ w2, const float* __restrict__ b2,
    const float* __restrict__ w3, const float* __restrict__ b3,
    float* __restrict__ recAcc) {
  __shared__ float sH[16*LDSW];   // 33 KB, conflict-free stride
  const int lane = threadIdx.x&31, wave = threadIdx.x>>5;
  const int row0 = blockIdx.x*16;
  const int mrow = lane&15, kh = (lane>>4)*2, ncol = lane&15, hl = lane>>4;
  const float* zrow = Z + (size_t)(row0+mrow)*64;

  v8f h2[8];
  { v8f zz = {}; for (int nt=0;nt<8;++nt) h2[nt] = zz; }

  // phase A: h1 (in 512-col halves through LDS) -> h2 accumulators in registers
  for (int half=0; half<2; ++half) {
    if (half) __syncthreads();
    for (int nt=0; nt<4; ++nt) {
      const int cn = wave*64 + nt*16;       // col within half
      const int gn = half*512 + cn;         // global h1 col
      v8f acc = {};
      for (int k0=0;k0<64;k0+=4){
        v2f a,bb;
        a.x  = zrow[k0+kh];                        a.y  = zrow[k0+kh+1];
        bb.x = w1[(size_t)(k0+kh)*1024 + gn+ncol]; bb.y = w1[(size_t)(k0+kh+1)*1024 + gn+ncol];
        acc = wmma4(a,bb,acc);
      }
      float bv = b1[gn+ncol];
      for (int r=0;r<8;++r)
        sH[(r+8*hl)*LDSW + cn + ncol] = fmaxf(acc[r]+bv, 0.f);
    }
    __syncthreads();
    for (int nt=0; nt<8; ++nt) {
      const int gn = wave*128 + nt*16;
      v8f acc = h2[nt];
      for (int k0=0;k0<512;k0+=4){
        v2f a,bb;
        a.x  = sH[mrow*LDSW + k0+kh];  a.y = sH[mrow*LDSW + k0+kh+1];
        bb.x = w2[(size_t)(half*512 + k0+kh)*1024 + gn+ncol];
        bb.y = w2[(size_t)(half*512 + k0+kh+1)*1024 + gn+ncol];
        acc = wmma4(a,bb,acc);
      }
      h2[nt] = acc;
    }
  }
  // bias + relu on h2
  for (int nt=0;nt<8;++nt) {
    float bv = b2[wave*128 + nt*16 + ncol];
    for (int r=0;r<8;++r) h2[nt][r] = fmaxf(h2[nt][r]+bv, 0.f);
  }
  // phase B: h3 = relu(h2@w3+b3) in 512-row K halves through LDS
  v8f h3[4];
  { v8f zz = {}; for (int nt=0;nt<4;++nt) h3[nt] = zz; }
  for (int half=0; half<2; ++half) {
    __syncthreads();
    if ((wave>>2) == half) {
      for (int nt=0;nt<8;++nt) {
        const int cn = (wave&3)*128 + nt*16;
        for (int r=0;r<8;++r)
          sH[(r+8*hl)*LDSW + cn + ncol] = h2[nt][r];
      }
    }
    __syncthreads();
    for (int nt=0;nt<4;++nt) {
      const int gn = wave*64 + nt*16;
      v8f acc = h3[nt];
      for (int k0=0;k0<512;k0+=4){
        v2f a,bb;
        a.x  = sH[mrow*LDSW + k0+kh];  a.y = sH[mrow*LDSW + k0+kh+1];
        bb.x = w3[(size_t)(half*512 + k0+kh)*512 + gn+ncol];
        bb.y = w3[(size_t)(half*512 + k0+kh+1)*512 + gn+ncol];
        acc = wmma4(a,bb,acc);
      }
      h3[nt] = acc;
    }
  }
  // rec loss
  float s = 0.f;
  for (int nt=0;nt<4;++nt) {
    const int col = wave*64 + nt*16 + ncol;
    float bv = b3[col];
    for (int r=0;r<8;++r) {
      const int row = row0 + r + 8*hl;
      float v = fmaxf(h3[nt][r]+bv, 0.f);
      float d = v - Hin[(size_t)(row/MC_)*D_ + col];
      s += d*d;
    }
  }
  s = wsum(s);
  if (lane==0) atomicAdd(recAcc, s);
}

// ---------------- KL term: log_qz - logsumexp_k(ll) per (b,s,mc) -----------------
__global__ __launch_bounds__(256) void kl_kernel(
    const float* __restrict__ stats, const float* __restrict__ eps,
    const float* __restrict__ Z, const float* __restrict__ x2,
    const float* __restrict__ mu, const float* __restrict__ m2, const float* __restrict__ lp,
    float* __restrict__ klAcc) {
  __shared__ float sMu[K_*64]; __shared__ float sM2[K_]; __shared__ float sLp[K_];
  const int lane = threadIdx.x&31, wave = threadIdx.x>>5;
  const int gw0 = blockIdx.x*8;
  const int b = gw0 / NALL_;
  for (int i=threadIdx.x; i<K_*64; i+=256) sMu[i] = mu[b*K_*64 + i];
  if (threadIdx.x < K_){ sM2[threadIdx.x]=m2[b*K_+threadIdx.x]; sLp[threadIdx.x]=lp[b*K_+threadIdx.x]; }
  __syncthreads();
  const int gw = gw0 + wave;
  const int bs = gw / MC_;
  float lv0 = stats[(size_t)bs*128 + 64 + lane];
  float lv1 = stats[(size_t)bs*128 + 96 + lane];
  float e0 = eps[(size_t)gw*64 + lane], e1 = eps[(size_t)gw*64 + lane+32];
  float q = -0.5f*(lv0 + e0*e0) - 0.5f*(lv1 + e1*e1);
  q = wsum(q) + C0_;
  float z0 = Z[(size_t)gw*64 + lane], z1 = Z[(size_t)gw*64 + lane+32];
  float ll[K_];
  #pragma unroll
  for (int k=0;k<K_;++k) {
    float p = z0*sMu[k*64+lane] + z1*sMu[k*64+lane+32];
    ll[k] = wsum(p);
  }
  float xv = x2[gw];
  float mx = -3.0e38f;
  #pragma unroll
  for (int k=0;k<K_;++k) {
    ll[k] = -0.5f*(xv - 2.f*ll[k] + sM2[k]) + C0_ + sLp[k];
    mx = fmaxf(mx, ll[k]);
  }
  float s = 0.f;
  #pragma unroll
  for (int k=0;k<K_;++k) s += expf(ll[k]-mx);
  float lse = mx + logf(s);
  if (lane==0) atomicAdd(klAcc, q - lse);
}

__global__ void zero_kernel(float* p, int n){
  int i = blockIdx.x*blockDim.x + threadIdx.x;
  if (i < n) p[i] = 0.f;
}
__global__ void finalize_kernel(const float* __restrict__ scal, float* __restrict__ out){
  out[0] = scal[0] / (float)ROWS_;
  out[1] = scal[1] / (float)ROWS_;
}

// =================================================================================
extern "C" void kernel_launch(void* const* d_in, const int* in_sizes, int n_in,
                              void* d_out, int out_size, void* d_ws, size_t ws_size,
                              hipStream_t stream) {
  (void)in_sizes; (void)n_in; (void)out_size; (void)ws_size;
  const float* H   = (const float*)d_in[0];
  const float* eps = (const float*)d_in[1];
  const float* wq  = (const float*)d_in[2];
  const float* bq  = (const float*)d_in[3];
  const float* wk  = (const float*)d_in[4];
  const float* bk  = (const float*)d_in[5];
  const float* wv  = (const float*)d_in[6];
  const float* bv  = (const float*)d_in[7];
  const float* wo  = (const float*)d_in[8];
  const float* bo  = (const float*)d_in[9];
  const float* wz  = (const float*)d_in[10];
  const float* bz  = (const float*)d_in[11];
  const float* w1  = (const float*)d_in[12];
  const float* b1  = (const float*)d_in[13];
  const float* w2  = (const float*)d_in[14];
  const float* b2  = (const float*)d_in[15];
  const float* w3  = (const float*)d_in[16];
  const float* b3  = (const float*)d_in[17];
  const int* init_idx = (const int*)d_in[18];

  float* ws = (float*)d_ws;
  size_t off = 0;
  auto alloc = [&](size_t n){ float* p = ws + off; off += n; return p; };
  float* hA    = alloc((size_t)BS_*D_);
  float* hB    = alloc((size_t)BS_*D_);
  float* qb    = alloc((size_t)BS_*D_);
  float* kb    = alloc((size_t)BS_*D_);
  float* vb    = alloc((size_t)BS_*D_);
  float* ob    = alloc((size_t)BS_*D_);
  float* att   = alloc((size_t)B_*NH_*S_*S_);
  float* stats = alloc((size_t)BS_*128);
  float* zb    = alloc((size_t)ROWS_*64);
  float* x2    = alloc((size_t)ROWS_);
  float* mu    = alloc((size_t)B_*K_*64);
  float* pi    = alloc((size_t)B_*K_);
  float* muAcc = alloc((size_t)B_*K_*64);   // contiguous with NkAcc for one zero pass
  float* NkAcc = alloc((size_t)B_*K_);
  float* m2    = alloc((size_t)B_*K_);
  float* lp    = alloc((size_t)B_*K_);
  float* scal  = alloc(2);                  // [0]=rec sum, [1]=kl sum

  zero_kernel<<<1,64,0,stream>>>(scal, 2);

  const float scale = 1.0f / sqrtf((float)D_);
  const float* cur = H;
  float* outH[2] = {hA, hB};
  for (int i=0;i<2;++i) {
    const float* Wq = wq + (size_t)i*D_*D_; const float* Bq = bq + (size_t)i*D_;
    const float* Wk = wk + (size_t)i*D_*D_; const float* Bk = bk + (size_t)i*D_;
    const float* Wv = wv + (size_t)i*D_*D_; const float* Bv = bv + (size_t)i*D_;
    const float* Wo = wo + (size_t)i*D_*D_; const float* Bo = bo + (size_t)i*D_;
    gemm_kernel<<<dim3(32,256),32,0,stream>>>(cur, Wq, Bq, nullptr, qb, D_, D_, 0);
    gemm_kernel<<<dim3(32,256),32,0,stream>>>(cur, Wk, Bk, nullptr, kb, D_, D_, 0);
    gemm_kernel<<<dim3(32,256),32,0,stream>>>(cur, Wv, Bv, nullptr, vb, D_, D_, 0);
    attn_scores_kernel<<<dim3(16,16,B_*NH_),32,0,stream>>>(qb, kb, att, scale);
    softmax_kernel<<<(B_*NH_*S_)/8,256,0,stream>>>(att);
    attn_v_kernel<<<dim3(8,16,B_*NH_),32,0,stream>>>(att, qb, vb, ob);
    gemm_kernel<<<dim3(32,256),32,0,stream>>>(ob, Wo, Bo, ob, outH[i], D_, D_, 2);
    cur = outH[i];
  }

  gemm_kernel<<<dim3(8,256),32,0,stream>>>(cur, wz, bz, nullptr, stats, 128, D_, 0);
  z_kernel<<<ROWS_/8,256,0,stream>>>(stats, eps, zb, x2);
  mu_init_kernel<<<dim3(B_,K_),64,0,stream>>>(zb, init_idx, mu, pi);

  for (int it=0; it<5; ++it) {
    zero_kernel<<<(B_*K_*64 + B_*K_ + 255)/256,256,0,stream>>>(muAcc, B_*K_*64 + B_*K_);
    em_prep_kernel<<<B_,32,0,stream>>>(mu, pi, m2, lp);
    em_accum_kernel<<<dim3(B_,NALL_/256),256,0,stream>>>(zb, x2, mu, m2, lp, muAcc, NkAcc);
    em_final_kernel<<<B_,64,0,stream>>>(muAcc, NkAcc, mu, pi);
  }
  em_prep_kernel<<<B_,32,0,stream>>>(mu, pi, m2, lp);   // m2/lp for final pi,mu

  decoder_kernel<<<ROWS_/16,256,0,stream>>>(zb, H, w1,b1,w2,b2,w3,b3, scal);
  kl_kernel<<<ROWS_/8,256,0,stream>>>(stats, eps, zb, x2, mu, m2, lp, scal+1);
  finalize_kernel<<<1,1,0,stream>>>(scal, (float*)d_out);
}